// VectorQuantizer_76424648065077
// MI455X (gfx1250) — compile-verified
//
#include <hip/hip_runtime.h>

typedef __attribute__((ext_vector_type(2))) float v2f;
typedef __attribute__((ext_vector_type(4))) float v4f;
typedef __attribute__((ext_vector_type(8))) float v8f;

#define VQ_N 262144
#define VQ_D 64
#define VQ_K 512

// Float index into the WMMA B-fragment image of embeddings[d][k] (D=64,K=512).
// Tile nt = k/16 (N), chunk kc = d/4 (K4). Lane: low 16 hold kk=0,1 ; high 16 kk=2,3.
__device__ __forceinline__ int flds(int d, int k) {
  int nt = k >> 4;
  int kc = d >> 2;
  int ln = ((d & 2) << 3) | (k & 15);       // +16 lanes when kk in {2,3}
  return ((((nt << 4) + kc) << 5) + ln) * 2 + (d & 1);
}

__global__ __launch_bounds__(256) void vq_kernel(const float* __restrict__ x,
                                                 const float* __restrict__ emb,
                                                 float* __restrict__ enc,
                                                 float* __restrict__ quant) {
  __shared__ float sFrag[VQ_D * VQ_K];      // 128 KB: B-fragment layout
  __shared__ float sEnorm[VQ_K];            // ||e_k||^2

  const int t = threadIdx.x;

  // ---- Stage embeddings into LDS (swizzled) + column squared norms.
  // iteration i covers row d=i/2, k = (i&1)*256 + t  (coalesced 1KB reads)
  float sq0 = 0.f, sq1 = 0.f;
  for (int i = 0; i < (VQ_D * VQ_K) / 256; ++i) {
    int idx = i * 256 + t;
    float v = emb[idx];
    int d = idx >> 9;
    int k = idx & 511;
    sFrag[flds(d, k)] = v;
    if (i & 1) sq1 += v * v; else sq0 += v * v;
  }
  sEnorm[t] = sq0;
  sEnorm[t + 256] = sq1;
  __syncthreads();

  const int wave = t >> 5;                  // 8 waves of 32 (wave32)
  const int lane = t & 31;
  const int l16  = lane & 15;
  const int hi   = lane >> 4;               // 0: kk 0,1 / rows 0-7 ; 1: kk 2,3 / rows 8-15
  const int row0 = (blockIdx.x * 8 + wave) * 16;

  // ---- A fragments: 16 rows x 64 cols of x, 16x4 f32 layout per chunk.
  const float* xrow = x + (size_t)(row0 + l16) * VQ_D + hi * 2;
  v2f a[16];
#pragma unroll
  for (int kc = 0; kc < 16; ++kc)
    a[kc] = *(const v2f*)(xrow + kc * 4);

  float bestVal[8];
  int   bestIdx[8];
#pragma unroll
  for (int v = 0; v < 8; ++v) { bestVal[v] = 3.4e38f; bestIdx[v] = 0; }

  // ---- similarity GEMM + running argmin of (||e||^2 - 2*sim); ||x||^2 is
  // constant per row so it never affects the argmin.
  const v2f* bbase = (const v2f*)sFrag;
  for (int nt = 0; nt < 32; ++nt) {
    v8f acc = {};
    const v2f* bp = bbase + (nt << 9) + lane;
#pragma unroll
    for (int kc = 0; kc < 16; ++kc) {
      v2f b = bp[kc << 5];                  // ds_load_b64
      acc = __builtin_amdgcn_wmma_f32_16x16x4_f32(false, a[kc], false, b,
                                                  (short)0, acc, false, false);
    }
    float en  = sEnorm[(nt << 4) + l16];
    int  kcur = (nt << 4) + l16;
#pragma unroll
    for (int v = 0; v < 8; ++v) {
      float score = __builtin_fmaf(-2.f, acc[v], en);
      bool lt = score < bestVal[v];         // branchless: keep EXEC uniform
      bestVal[v] = lt ? score : bestVal[v];
      bestIdx[v] = lt ? kcur  : bestIdx[v];
    }
  }

  // ---- Reduce over the 16-lane N dimension (xor 1,2,4,8 stays inside halves).
  // First-index wins on ties, matching jnp.argmin.
#pragma unroll
  for (int m = 1; m <= 8; m <<= 1) {
#pragma unroll
    for (int v = 0; v < 8; ++v) {
      float ov = __shfl_xor(bestVal[v], m, 32);
      int   oi = __shfl_xor(bestIdx[v], m, 32);
      bool take = (ov < bestVal[v]) || (ov == bestVal[v] && oi < bestIdx[v]);
      bestVal[v] = take ? ov : bestVal[v];
      bestIdx[v] = take ? oi : bestIdx[v];
    }
  }

  // ---- Outputs. Low half lanes write rows row0+0..7, high half rows row0+8..15.
  // Per row: 16 lanes cover contiguous 256B segments -> coalesced NT streams.
#pragma unroll
  for (int v = 0; v < 8; ++v) {
    int row = row0 + hi * 8 + v;
    int kb  = bestIdx[v];                   // uniform within each half

    // quantized[row][:] = embeddings[:, kb]  (gather from LDS fragment image)
    v2f q01 = *(const v2f*)&sFrag[flds(l16 * 4 + 0, kb)];
    v2f q23 = *(const v2f*)&sFrag[flds(l16 * 4 + 2, kb)];
    v4f q; q[0] = q01[0]; q[1] = q01[1]; q[2] = q23[0]; q[3] = q23[1];
    __builtin_nontemporal_store(q, (v4f*)(quant + (size_t)row * VQ_D + l16 * 4));

    // encodings[row][:] = one_hot(kb)
    float* erow = enc + (size_t)row * VQ_K;
#pragma unroll
    for (int it = 0; it < 8; ++it) {
      int k0 = it * 64 + l16 * 4;
      v4f e;
      e[0] = (k0 + 0 == kb) ? 1.f : 0.f;
      e[1] = (k0 + 1 == kb) ? 1.f : 0.f;
      e[2] = (k0 + 2 == kb) ? 1.f : 0.f;
      e[3] = (k0 + 3 == kb) ? 1.f : 0.f;
      __builtin_nontemporal_store(e, (v4f*)(erow + k0));
    }
  }
}

extern "C" void kernel_launch(void* const* d_in, const int* in_sizes, int n_in,
                              void* d_out, int out_size, void* d_ws, size_t ws_size,
                              hipStream_t stream) {
  const float* x   = (const float*)d_in[0];     // [N, D] f32
  const float* emb = (const float*)d_in[1];     // [D, K] f32
  float* enc   = (float*)d_out;                 // [N, K] f32
  float* quant = (float*)d_out + (size_t)VQ_N * VQ_K;  // [N, D] f32
  dim3 grid(VQ_N / 128);                        // 128 rows per 256-thread block
  vq_kernel<<<grid, 256, 0, stream>>>(x, emb, enc, quant);
}